// align_gcn_16020228014505
// MI455X (gfx1250) — compile-verified
//
#include <hip/hip_runtime.h>

typedef __attribute__((ext_vector_type(2))) float v2f;
typedef __attribute__((ext_vector_type(8))) float v8f;

#define D 256
#define PANELS 4           // four 64-wide column panels per row of output
                           // (each wave: 16 rows x 64 cols = 4 WMMA tiles)

// ---------------------------------------------------------------------------
// Kernel 1: h = X @ W1  (f32 WMMA, one 16x64 strip per wave32, 4x A-reuse)
// ---------------------------------------------------------------------------
__global__ void __launch_bounds__(128)
gemm_h_wmma(const float* __restrict__ X, const float* __restrict__ W,
            float* __restrict__ H) {
    const int wave  = blockIdx.x * (blockDim.x >> 5) + (threadIdx.x >> 5);
    const int lane  = threadIdx.x & 31;
    const int tm    = wave >> 2;            // row tile   [0, 6250)
    const int panel = wave & 3;             // col panel  [0, 4): 64 columns
    const int half  = lane >> 4;            // 0: lanes 0-15, 1: lanes 16-31
    const int idx   = lane & 15;

    const int rowA    = tm * 16 + idx;      // A: M = idx (both halves)
    const int colBase = panel * 64 + idx;   // B: N = idx + t*16

    v8f acc0 = {}, acc1 = {}, acc2 = {}, acc3 = {};
    const float* __restrict__ arow = X + (size_t)rowA * D;

    #pragma unroll 4
    for (int k = 0; k < D; k += 4) {
        // A 16x4 f32: lane holds K = {k+2*half, k+2*half+1} (contiguous b64)
        v2f a = *(const v2f*)(arow + k + 2 * half);
        // B rows K = k+half (VGPR0) and K = k+2+half (VGPR1), N = colBase+t*16
        const float* __restrict__ w0 = W + (size_t)(k + half)     * D + colBase;
        const float* __restrict__ w1 = W + (size_t)(k + 2 + half) * D + colBase;
        v2f b0 = {w0[0],  w1[0]};
        v2f b1 = {w0[16], w1[16]};
        v2f b2 = {w0[32], w1[32]};
        v2f b3 = {w0[48], w1[48]};
        acc0 = __builtin_amdgcn_wmma_f32_16x16x4_f32(false, a, false, b0, (short)0, acc0, false, false);
        acc1 = __builtin_amdgcn_wmma_f32_16x16x4_f32(false, a, false, b1, (short)0, acc1, false, false);
        acc2 = __builtin_amdgcn_wmma_f32_16x16x4_f32(false, a, false, b2, (short)0, acc2, false, false);
        acc3 = __builtin_amdgcn_wmma_f32_16x16x4_f32(false, a, false, b3, (short)0, acc3, false, false);
    }

    // C/D 16x16 f32: vgpr r -> (M = r + 8*half, N = idx); 4 tiles side by side
    #pragma unroll
    for (int r = 0; r < 8; ++r) {
        float* __restrict__ orow = H + (size_t)(tm * 16 + r + 8 * half) * D + panel * 64 + idx;
        orow[0]  = acc0[r];
        orow[16] = acc1[r];
        orow[32] = acc2[r];
        orow[48] = acc3[r];
    }
}

// ---------------------------------------------------------------------------
// Kernel 2: segmented SpMM. edge_rows is sorted, so each block owns one output
// row: binary-search its edge span, accumulate privately (no atomics), store
// pre-ReLU agg. Thread = one of D=256 columns -> coalesced 1KB row gathers;
// prefetch the next gathered row to hide latency (global_prefetch_b8).
// ---------------------------------------------------------------------------
__global__ void __launch_bounds__(256)
spmm_rows(const float* __restrict__ H, const float* __restrict__ vals,
          const int* __restrict__ rows, const int* __restrict__ cols,
          float* __restrict__ agg, int E) {
    const int r = blockIdx.x;
    __shared__ int s_lo, s_hi;
    if (threadIdx.x == 0) {
        int lo = 0, hi = E;                       // lower_bound(rows, r)
        while (lo < hi) { int m = (lo + hi) >> 1; if (rows[m] < r) lo = m + 1; else hi = m; }
        s_lo = lo;
        hi = E;                                   // lower_bound(rows, r+1)
        while (lo < hi) { int m = (lo + hi) >> 1; if (rows[m] < r + 1) lo = m + 1; else hi = m; }
        s_hi = lo;
    }
    __syncthreads();
    const int lo = s_lo, hi = s_hi;
    const int d = threadIdx.x;

    float acc = 0.0f;
    for (int e = lo; e < hi; ++e) {
        const int   c = cols[e];
        const float v = vals[e];
        if (e + 1 < hi) {
            __builtin_prefetch(H + (size_t)cols[e + 1] * D + d, 0, 1);
        }
        acc = fmaf(v, H[(size_t)c * D + d], acc);
    }
    agg[(size_t)r * D + d] = acc;   // pre-ReLU; ReLU fused into kernel 3
}

// ---------------------------------------------------------------------------
// Kernel 3: gate = sigmoid(X[perm] @ Wr + br); out = gate*relu(agg)+(1-gate)*left
// Same 16x64 register blocking; the perm-gathered A row fetch is amortized
// across 4 WMMA tiles. agg read from d_out, result written back in place.
// ---------------------------------------------------------------------------
__global__ void __launch_bounds__(128)
highway_wmma(const float* __restrict__ X, const float* __restrict__ Wr,
             const float* __restrict__ br, const int* __restrict__ perm,
             float* __restrict__ out) {
    const int wave  = blockIdx.x * (blockDim.x >> 5) + (threadIdx.x >> 5);
    const int lane  = threadIdx.x & 31;
    const int tm    = wave >> 2;
    const int panel = wave & 3;
    const int half  = lane >> 4;
    const int idx   = lane & 15;

    const int m0      = tm * 16;
    const int colBase = panel * 64 + idx;
    const int pA      = perm[m0 + idx];           // gathered A row (M = idx)
    const float* __restrict__ arow = X + (size_t)pA * D;

    v8f acc0 = {}, acc1 = {}, acc2 = {}, acc3 = {};
    #pragma unroll 4
    for (int k = 0; k < D; k += 4) {
        v2f a = *(const v2f*)(arow + k + 2 * half);
        const float* __restrict__ w0 = Wr + (size_t)(k + half)     * D + colBase;
        const float* __restrict__ w1 = Wr + (size_t)(k + 2 + half) * D + colBase;
        v2f b0 = {w0[0],  w1[0]};
        v2f b1 = {w0[16], w1[16]};
        v2f b2 = {w0[32], w1[32]};
        v2f b3 = {w0[48], w1[48]};
        acc0 = __builtin_amdgcn_wmma_f32_16x16x4_f32(false, a, false, b0, (short)0, acc0, false, false);
        acc1 = __builtin_amdgcn_wmma_f32_16x16x4_f32(false, a, false, b1, (short)0, acc1, false, false);
        acc2 = __builtin_amdgcn_wmma_f32_16x16x4_f32(false, a, false, b2, (short)0, acc2, false, false);
        acc3 = __builtin_amdgcn_wmma_f32_16x16x4_f32(false, a, false, b3, (short)0, acc3, false, false);
    }

    const float bias0 = br[colBase +  0];
    const float bias1 = br[colBase + 16];
    const float bias2 = br[colBase + 32];
    const float bias3 = br[colBase + 48];

    #pragma unroll
    for (int r = 0; r < 8; ++r) {
        const int m = m0 + r + 8 * half;
        const size_t rbase = (size_t)m * D + panel * 64 + idx;
        const float* __restrict__ lrow = X + (size_t)perm[m] * D + panel * 64 + idx;

        float pre[4]  = {acc0[r] + bias0, acc1[r] + bias1,
                         acc2[r] + bias2, acc3[r] + bias3};
        #pragma unroll
        for (int t = 0; t < 4; ++t) {
            const float g    = 1.0f / (1.0f + __expf(-pre[t]));
            const float left = lrow[t * 16];
            const float a    = out[rbase + t * 16];        // agg from kernel 2
            const float relu = a > 0.0f ? a : 0.0f;
            out[rbase + t * 16] = g * relu + (1.0f - g) * left;
        }
    }
}

// ---------------------------------------------------------------------------
extern "C" void kernel_launch(void* const* d_in, const int* in_sizes, int n_in,
                              void* d_out, int out_size, void* d_ws, size_t ws_size,
                              hipStream_t stream) {
    const float* X     = (const float*)d_in[0];   // right_embed [N,D]
    const float* W1    = (const float*)d_in[1];   // [D,D]
    const float* Wr    = (const float*)d_in[2];   // [D,D]
    const float* br    = (const float*)d_in[3];   // [D]
    const float* evals = (const float*)d_in[4];   // [E]
    const int*   erows = (const int*)d_in[5];     // [E] sorted
    const int*   ecols = (const int*)d_in[6];     // [E]
    const int*   perm  = (const int*)d_in[7];     // [N]

    const int N = in_sizes[7];
    const int E = in_sizes[4];

    float* h   = (float*)d_ws;                    // [N,D] scratch (102.4 MB)
    float* out = (float*)d_out;                   // [N,D]; holds agg then result

    const int waves  = (N / 16) * PANELS;         // 16x64 strip per wave
    const int blocks = waves / 4;                 // 128 threads = 4 waves/block

    gemm_h_wmma <<<blocks, 128, 0, stream>>>(X, W1, h);
    spmm_rows   <<<N,      256, 0, stream>>>(h, evals, erows, ecols, out, E);
    highway_wmma<<<blocks, 128, 0, stream>>>(X, Wr, br, perm, out);
}